// BaselineKVCacheAttention_11003706212930
// MI455X (gfx1250) — compile-verified
//
#include <hip/hip_runtime.h>
#include <hip/hip_bf16.h>
#include <math.h>

// ---------------------------------------------------------------------------
// KV-cache attention for MI455X (gfx1250, wave32).
// Memory-bound: ~1.07 GB cache reads + ~1.07 GB concat-output writes dominate
// (23.3 TB/s -> ~92 us floor). The cache copy is FUSED into the QK^T and PV
// kernels so each cache byte is read exactly once, staged through LDS with
// coalesced B128 traffic (async global->LDS: global_load_async_to_lds_b128 +
// s_wait_asynccnt), and consumed as V_WMMA_F32_16X16X4_F32 operands from LDS.
// ---------------------------------------------------------------------------

typedef __attribute__((ext_vector_type(2))) float v2f;
typedef __attribute__((ext_vector_type(8))) float v8f;
typedef int v4i_ __attribute__((vector_size(16)));   // pointee type the async builtin wants

#define B_     16
#define S_     4
#define H_     1024
#define NH_    16
#define CACHE_ 8192
#define HD_    64
#define H3_    3072
#define KV_ROWS_ (CACHE_ + S_)   // 8196
#define BS_    (B_ * S_)         // 64
#define PITCH_ 68                // LDS row pitch (floats): 16B-aligned, bank-spread

#ifndef __has_builtin
#define __has_builtin(x) 0
#endif
#if __has_builtin(__builtin_amdgcn_global_load_async_to_lds_b128)
#define HAVE_ASYNC_LDS 1
#else
#define HAVE_ASYNC_LDS 0
#endif

#if HAVE_ASYNC_LDS
// (global v4i*, lds v4i*, imm offset, imm cpol) — confirmed 4 args by frontend.
#define ASG(p) ((__attribute__((address_space(1))) v4i_*)(uintptr_t)(p))
#define ASL(p) ((__attribute__((address_space(3))) v4i_*)(uint32_t)(uintptr_t)(p))
__device__ __forceinline__ void async_wait0() {
#if __has_builtin(__builtin_amdgcn_s_wait_asynccnt)
  __builtin_amdgcn_s_wait_asynccnt(0);
#else
  asm volatile("s_wait_asynccnt 0x0" ::: "memory");
#endif
}
#endif

// ---------------------------------------------------------------------------
// D = A[M x K] * W[K x N] + bias. One wave per 16x16 tile (tiny GEMMs).
template <int K, int N>
__global__ void gemm_wmma(const float* __restrict__ A,
                          const float* __restrict__ W,
                          const float* __restrict__ bias,
                          float* __restrict__ out) {
  const int lane = threadIdx.x;
  const int half = lane >> 4;
  const int l    = lane & 15;
  const int n0 = blockIdx.x * 16;
  const int m0 = blockIdx.y * 16;

  v8f c = {};
  for (int k = 0; k < K; k += 4) {
    const int kk = k + 2 * half;
    v2f a = *(const v2f*)(A + (size_t)(m0 + l) * K + kk);
    v2f b;
    b.x = W[(size_t)(kk)     * N + n0 + l];
    b.y = W[(size_t)(kk + 1) * N + n0 + l];
    c = __builtin_amdgcn_wmma_f32_16x16x4_f32(false, a, false, b,
                                              (short)0, c, false, false);
  }
  const float bb = bias[n0 + l];
#pragma unroll
  for (int i = 0; i < 8; ++i) {
    const int row = m0 + i + 8 * half;   // VGPR i: M=i (lanes 0-15), M=8+i (16-31)
    out[(size_t)row * N + n0 + l] = c[i] + bb;
  }
}

// ---------------------------------------------------------------------------
// Scatter the S_ freshly projected K/V rows into the grown-cache outputs.
__global__ void scatter_new_kv(const float* __restrict__ qkv,
                               float* __restrict__ k_out,
                               float* __restrict__ v_out) {
  int idx = blockIdx.x * blockDim.x + threadIdx.x;   // B*NH*S*HD = 65536
  if (idx >= B_ * NH_ * S_ * HD_) return;
  const int d = idx & (HD_ - 1);
  const int s = (idx >> 6) & (S_ - 1);
  const int h = (idx >> 8) & (NH_ - 1);
  const int b = idx >> 12;
  const int t = b * S_ + s;
  const size_t ro = ((size_t)(b * NH_ + h) * KV_ROWS_ + CACHE_ + s) * HD_ + d;
  k_out[ro] = qkv[(size_t)t * H3_ + H_     + h * HD_ + d];
  v_out[ro] = qkv[(size_t)t * H3_ + 2 * H_ + h * HD_ + d];
}

// Scores against the S_ new K rows (tiny: 4096 dot-64s).
__global__ void score_new(const float* __restrict__ qkv,
                          float* __restrict__ scores) {
  int idx = blockIdx.x * blockDim.x + threadIdx.x;   // B*NH*S*S = 4096
  if (idx >= B_ * NH_ * S_ * S_) return;
  const int s = idx & 3;
  const int q = (idx >> 2) & 3;
  const int h = (idx >> 4) & 15;
  const int b = idx >> 8;
  const float* qv = qkv + (size_t)(b * S_ + q) * H3_ + h * HD_;
  const float* kv = qkv + (size_t)(b * S_ + s) * H3_ + H_ + h * HD_;
  float acc = 0.f;
#pragma unroll 8
  for (int d = 0; d < HD_; ++d) acc += qv[d] * kv[d];
  scores[((size_t)(b * NH_ + h) * S_ + q) * KV_ROWS_ + CACHE_ + s] = acc * 0.125f;
}

// ---------------------------------------------------------------------------
// Q @ K^T fused with k_cache -> k_out copy. Block = 256 threads (8 waves),
// 128 K-rows per block. K chunk staged to LDS with coalesced B128 traffic
// (async global->LDS), copied out with B128 stores, then each wave contracts
// its 16-row tile via WMMA with LDS-resident operands.
__global__ void scores_k_wmma(const float* __restrict__ qkv,
                              const float* __restrict__ k_cache,
                              float* __restrict__ k_out,
                              float* __restrict__ scores) {
  __shared__ __align__(16) float qs[S_ * PITCH_];
  __shared__ __align__(16) float ks[128 * PITCH_];
  const int t = threadIdx.x;               // 0..255
  const int row0 = blockIdx.x * 128;
  const int bh = blockIdx.y;
  const int b = bh >> 4, h = bh & 15;

  // stage q tile (4 x 64)
  {
    const int q = t >> 6, d = t & 63;
    qs[q * PITCH_ + d] = qkv[(size_t)(b * S_ + q) * H3_ + h * HD_ + d];
  }
  // stage K chunk: 128 x 64 floats = 2048 float4, 8 per thread, coalesced
  const size_t cbase = (size_t)bh * CACHE_ * HD_;
#pragma unroll
  for (int r = 0; r < 8; ++r) {
    const int i = t + 256 * r;
    const int row = i >> 4, c4 = (i & 15) * 4;
    const float* gp = k_cache + cbase + (size_t)(row0 + row) * HD_ + c4;
    float* lp = &ks[row * PITCH_ + c4];
#if HAVE_ASYNC_LDS
    __builtin_amdgcn_global_load_async_to_lds_b128(ASG(gp), ASL(lp), 0, 0);
#else
    *(float4*)lp = *(const float4*)gp;
#endif
  }
#if HAVE_ASYNC_LDS
  async_wait0();
#endif
  __syncthreads();

  // fused copy: chunk -> k_out with coalesced B128 stores
  const size_t obase = (size_t)bh * KV_ROWS_ * HD_;
#pragma unroll
  for (int r = 0; r < 8; ++r) {
    const int i = t + 256 * r;
    const int row = i >> 4, c4 = (i & 15) * 4;
    float4 v = *(const float4*)&ks[row * PITCH_ + c4];
    *(float4*)(k_out + obase + (size_t)(row0 + row) * HD_ + c4) = v;
  }

  // WMMA: wave w owns rows [row0 + w*16, +16)
  const int w = t >> 5, lane = t & 31, half = lane >> 4, l = lane & 15;
  const float amask = (l < S_) ? 1.f : 0.f;    // q has only 4 real rows
  const int arow = (l < S_) ? l : 0;
  v8f c = {};
#pragma unroll
  for (int j = 0; j < HD_ / 4; ++j) {
    const int kk = 4 * j + 2 * half;
    v2f a = *(const v2f*)&qs[arow * PITCH_ + kk];
    a.x *= amask; a.y *= amask;
    v2f bm = *(const v2f*)&ks[(w * 16 + l) * PITCH_ + kk];
    c = __builtin_amdgcn_wmma_f32_16x16x4_f32(false, a, false, bm,
                                              (short)0, c, false, false);
  }
  if (half == 0) {
    const int n = row0 + w * 16 + l;
#pragma unroll
    for (int i = 0; i < S_; ++i)
      scores[((size_t)bh * S_ + i) * KV_ROWS_ + n] = c[i] * 0.125f;
  }
}

// ---------------------------------------------------------------------------
// Row softmax over KV_ROWS_ (in place in workspace).
__global__ void softmax_rows(float* __restrict__ scores) {
  float* p = scores + (size_t)blockIdx.x * KV_ROWS_;
  __shared__ float red[256];
  const int t = threadIdx.x;

  float m = -INFINITY;
  for (int i = t; i < KV_ROWS_; i += 256) m = fmaxf(m, p[i]);
  red[t] = m; __syncthreads();
  for (int s = 128; s > 0; s >>= 1) {
    if (t < s) red[t] = fmaxf(red[t], red[t + s]);
    __syncthreads();
  }
  m = red[0]; __syncthreads();

  float sum = 0.f;
  for (int i = t; i < KV_ROWS_; i += 256) {
    float e = __expf(p[i] - m);
    p[i] = e;
    sum += e;
  }
  red[t] = sum; __syncthreads();
  for (int s = 128; s > 0; s >>= 1) {
    if (t < s) red[t] += red[t + s];
    __syncthreads();
  }
  const float inv = 1.0f / red[0];
  for (int i = t; i < KV_ROWS_; i += 256) p[i] *= inv;
}

// ---------------------------------------------------------------------------
// P @ V fused with v_cache -> v_out copy. Grid (8 row-chunks x 256 bh),
// block = 128 threads (4 waves, one per 16-wide head-dim tile). Each 64-row
// sub-chunk is staged to LDS (coalesced B128, async), streamed to v_out, then
// contracted via WMMA. Partials land in ws; the new V rows are folded in by
// chunk 7 under a *uniform* branch (EXEC stays all-ones).
__global__ void av_v_wmma(const float* __restrict__ attn,
                          const float* __restrict__ v_cache,
                          float* __restrict__ v_out,
                          float* __restrict__ part) {
  __shared__ __align__(16) float vs[64 * PITCH_];
  __shared__ __align__(16) float as_[S_ * PITCH_];
  const int t = threadIdx.x;               // 0..127
  const int w = t >> 5, lane = t & 31, half = lane >> 4, l = lane & 15;
  const int bh = blockIdx.y;
  const int ch = blockIdx.x;               // 0..7, 1024 rows each
  const int col = w * 16 + l;
  const size_t vcb = (size_t)bh * CACHE_   * HD_;
  const size_t vob = (size_t)bh * KV_ROWS_ * HD_;
  const float* arows = attn + (size_t)bh * S_ * KV_ROWS_;
  const float amask = (l < S_) ? 1.f : 0.f;
  const int arow = (l < S_) ? l : 0;

  v8f c = {};
  for (int s64 = 0; s64 < 16; ++s64) {
    const int r0 = ch * 1024 + s64 * 64;
    // stage attn 4 x 64 for this sub-chunk
    {
      const int i0 = t, i1 = t + 128;
      as_[(i0 >> 6) * PITCH_ + (i0 & 63)] =
          arows[(size_t)(i0 >> 6) * KV_ROWS_ + r0 + (i0 & 63)];
      as_[(i1 >> 6) * PITCH_ + (i1 & 63)] =
          arows[(size_t)(i1 >> 6) * KV_ROWS_ + r0 + (i1 & 63)];
    }
    // stage V 64 x 64 = 1024 float4, 8 per thread, coalesced
#pragma unroll
    for (int r = 0; r < 8; ++r) {
      const int i = t + 128 * r;
      const int row = i >> 4, c4 = (i & 15) * 4;
      const float* gp = v_cache + vcb + (size_t)(r0 + row) * HD_ + c4;
      float* lp = &vs[row * PITCH_ + c4];
#if HAVE_ASYNC_LDS
      __builtin_amdgcn_global_load_async_to_lds_b128(ASG(gp), ASL(lp), 0, 0);
#else
      *(float4*)lp = *(const float4*)gp;
#endif
    }
#if HAVE_ASYNC_LDS
    async_wait0();
#endif
    __syncthreads();
    // fused copy: sub-chunk -> v_out
#pragma unroll
    for (int r = 0; r < 8; ++r) {
      const int i = t + 128 * r;
      const int row = i >> 4, c4 = (i & 15) * 4;
      float4 v = *(const float4*)&vs[row * PITCH_ + c4];
      *(float4*)(v_out + vob + (size_t)(r0 + row) * HD_ + c4) = v;
    }
    // WMMA over the 64 rows (16 steps of K=4)
#pragma unroll
    for (int j = 0; j < 16; ++j) {
      const int kk = 4 * j + 2 * half;
      v2f a = *(const v2f*)&as_[arow * PITCH_ + kk];
      a.x *= amask; a.y *= amask;
      v2f bm;
      bm.x = vs[(kk)     * PITCH_ + col];
      bm.y = vs[(kk + 1) * PITCH_ + col];
      c = __builtin_amdgcn_wmma_f32_16x16x4_f32(false, a, false, bm,
                                                (short)0, c, false, false);
    }
    __syncthreads();
  }
  if (ch == 7) {   // uniform branch: fold in the 4 appended V rows
    const int kk = CACHE_ + 2 * half;
    v2f a;
    a.x = arows[(size_t)arow * KV_ROWS_ + kk]     * amask;
    a.y = arows[(size_t)arow * KV_ROWS_ + kk + 1] * amask;
    v2f bm;
    bm.x = v_out[vob + (size_t)(kk)     * HD_ + col];
    bm.y = v_out[vob + (size_t)(kk + 1) * HD_ + col];
    c = __builtin_amdgcn_wmma_f32_16x16x4_f32(false, a, false, bm,
                                              (short)0, c, false, false);
  }
  if (half == 0) {
#pragma unroll
    for (int i = 0; i < S_; ++i)   // q rows 0..3
      part[((size_t)ch * 256 + bh) * 256 + i * 64 + col] = c[i];
  }
}

// Deterministic reduction of the 8 row-chunk partials into x2 [64,1024].
__global__ void reduce_partials(const float* __restrict__ part,
                                float* __restrict__ x2) {
  int idx = blockIdx.x * blockDim.x + threadIdx.x;   // 65536
  if (idx >= B_ * NH_ * S_ * HD_) return;
  const int d = idx & 63;
  const int q = (idx >> 6) & 3;
  const int bh = idx >> 8;
  float s = 0.f;
#pragma unroll
  for (int ch = 0; ch < 8; ++ch)
    s += part[((size_t)ch * 256 + bh) * 256 + q * 64 + d];
  const int b = bh >> 4, h = bh & 15;
  x2[(size_t)(b * S_ + q) * H_ + h * HD_ + d] = s;
}

// ---------------------------------------------------------------------------
extern "C" void kernel_launch(void* const* d_in, const int* in_sizes, int n_in,
                              void* d_out, int out_size, void* d_ws, size_t ws_size,
                              hipStream_t stream) {
  const float* x       = (const float*)d_in[0];
  const float* k_cache = (const float*)d_in[1];
  const float* v_cache = (const float*)d_in[2];
  const float* W_in    = (const float*)d_in[3];
  const float* b_in    = (const float*)d_in[4];
  const float* W_out   = (const float*)d_in[5];
  const float* b_out   = (const float*)d_in[6];

  float* out   = (float*)d_out;                              // [B,S,H]
  float* k_out = out + (size_t)BS_ * H_;                     // [B,NH,8196,HD]
  float* v_out = k_out + (size_t)B_ * NH_ * KV_ROWS_ * HD_;  // [B,NH,8196,HD]

  float* ws     = (float*)d_ws;
  float* qkv    = ws;                                   // 196608 f
  float* x2     = qkv + (size_t)BS_ * H3_;              // 65536 f
  float* scores = x2 + (size_t)BS_ * H_;                // 8392704 f (~33.6 MB)
  float* part   = scores + (size_t)B_ * NH_ * S_ * KV_ROWS_;  // 524288 f (2 MB)

  // 1) QKV projection: [64,1024] @ [1024,3072]
  gemm_wmma<H_, H3_><<<dim3(H3_ / 16, BS_ / 16), 32, 0, stream>>>(x, W_in, b_in, qkv);
  // 2) place new K/V rows into grown-cache outputs; new-row scores
  scatter_new_kv<<<(B_ * NH_ * S_ * HD_) / 256, 256, 0, stream>>>(qkv, k_out, v_out);
  score_new<<<(B_ * NH_ * S_ * S_) / 256, 256, 0, stream>>>(qkv, scores);
  // 3) Q@K^T fused with K-cache copy (cache read exactly once, LDS-staged)
  scores_k_wmma<<<dim3(CACHE_ / 128, B_ * NH_), 256, 0, stream>>>(qkv, k_cache, k_out, scores);
  // 4) softmax over 8196 per (b,h,q)
  softmax_rows<<<B_ * NH_ * S_, 256, 0, stream>>>(scores);
  // 5) P@V fused with V-cache copy (LDS-staged, chunked partials)
  av_v_wmma<<<dim3(8, B_ * NH_), 128, 0, stream>>>(scores, v_cache, v_out, part);
  reduce_partials<<<(B_ * NH_ * S_ * HD_) / 256, 256, 0, stream>>>(part, x2);
  // 6) output projection: [64,1024] @ [1024,1024]
  gemm_wmma<H_, H_><<<dim3(H_ / 16, BS_ / 16), 32, 0, stream>>>(x2, W_out, b_out, out);
}